// FamilyBilinearAttention_4853313044700
// MI455X (gfx1250) — compile-verified
//
#include <hip/hip_runtime.h>
#include <hip/hip_bf16.h>
#include <math.h>

// Problem constants (from reference)
#define BB   256   // batches
#define NN   512   // nodes per graph
#define DG   256   // DIM_GRAPH
#define DD   256   // DIM_DESC
#define RK   64    // RANK
#define WAVES 8    // 256 threads / wave32

typedef float v2f __attribute__((ext_vector_type(2)));
typedef float v8f __attribute__((ext_vector_type(8)));

// ---------------------------------------------------------------------------
// Kernel 1: one block per batch. Computes w~ = (Wk^T q)/scale, streams H once,
// online softmax + alpha-weighted H accumulation, writes alpha and hbar.
// ---------------------------------------------------------------------------
__global__ __launch_bounds__(256) void attn_stream_kernel(
    const float* __restrict__ tokns,   // [B, DD]
    const float* __restrict__ H,       // [B, N, DG]
    const unsigned char* __restrict__ mask, // [B, N] (bool)
    const float* __restrict__ Wq,      // [RK, DD]
    const float* __restrict__ Wk,      // [RK, DG]
    const float* __restrict__ log_scale, // [1]
    float* __restrict__ alpha_out,     // [B, N]
    float* __restrict__ hbar_out)      // [B, DG] workspace
{
    const int b    = blockIdx.x;
    const int t    = threadIdx.x;
    const int wave = t >> 5;
    const int lane = t & 31;

    __shared__ __align__(16) float q_lds[RK];
    __shared__ __align__(16) float wt_lds[DG];
    __shared__ __align__(16) float s_lds[NN];
    __shared__ float m_lds[WAVES];
    __shared__ float z_lds[WAVES];
    __shared__ __align__(16) float hbar[DG];

    hbar[t] = 0.0f;

    // ---- q[r] = tokns[b] . Wq[r]  (tiny, L2-resident weights) ----
    if (t < RK) {
        const float* tk = tokns + (size_t)b * DD;
        const float* wq = Wq + (size_t)t * DD;
        float acc = 0.0f;
        #pragma unroll 8
        for (int d = 0; d < DD; ++d) acc = fmaf(tk[d], wq[d], acc);
        q_lds[t] = acc;
    }
    __syncthreads();

    // ---- w~[d] = (sum_r q[r] * Wk[r][d]) / clamp(exp(log_scale),0.1) ----
    {
        const float scale = fmaxf(__expf(log_scale[0]), 0.1f);
        float acc = 0.0f;
        #pragma unroll
        for (int r = 0; r < RK; ++r)
            acc = fmaf(q_lds[r], Wk[(size_t)r * DG + t], acc);
        wt_lds[t] = acc / scale;
    }
    __syncthreads();

    // per-lane slice of w~ : columns lane*8 .. lane*8+7
    const float4 w0 = *(const float4*)&wt_lds[lane * 8];
    const float4 w1 = *(const float4*)&wt_lds[lane * 8 + 4];

    // ---- single pass over this wave's 64 rows of H[b] ----
    float m = -INFINITY, z = 0.0f;
    float4 a0 = {0.f, 0.f, 0.f, 0.f};
    float4 a1 = {0.f, 0.f, 0.f, 0.f};
    const float* Hb = H + (size_t)b * NN * DG;
    const unsigned char* mb = mask + (size_t)b * NN;

    for (int i = 0; i < NN / WAVES; ++i) {
        const int n = wave * (NN / WAVES) + i;
        const float* rp = Hb + (size_t)n * DG + lane * 8;
        // prefetch next row's cacheline for this lane (global_prefetch_b8)
        __builtin_prefetch(rp + DG, 0, 0);
        const float4 h0 = *(const float4*)rp;
        const float4 h1 = *(const float4*)(rp + 4);

        // partial dot: H[n, lane*8 + j] * w~[lane*8 + j]
        float p = 0.0f;
        p = fmaf(h0.x, w0.x, p); p = fmaf(h0.y, w0.y, p);
        p = fmaf(h0.z, w0.z, p); p = fmaf(h0.w, w0.w, p);
        p = fmaf(h1.x, w1.x, p); p = fmaf(h1.y, w1.y, p);
        p = fmaf(h1.z, w1.z, p); p = fmaf(h1.w, w1.w, p);
        // wave32 cross-lane reduction -> full score in every lane
        #pragma unroll
        for (int off = 16; off > 0; off >>= 1)
            p += __shfl_xor(p, off, 32);

        const float sc = mb[n] ? p : -INFINITY;
        if (lane == 0) s_lds[n] = sc;

        // online softmax update (flash-style)
        const float mn   = fmaxf(m, sc);
        const float cold = (mn == -INFINITY) ? 1.0f : __expf(m - mn);
        const float w    = (sc == -INFINITY) ? 0.0f : __expf(sc - mn);
        z = z * cold + w;
        a0.x = fmaf(w, h0.x, a0.x * cold); a0.y = fmaf(w, h0.y, a0.y * cold);
        a0.z = fmaf(w, h0.z, a0.z * cold); a0.w = fmaf(w, h0.w, a0.w * cold);
        a1.x = fmaf(w, h1.x, a1.x * cold); a1.y = fmaf(w, h1.y, a1.y * cold);
        a1.z = fmaf(w, h1.z, a1.z * cold); a1.w = fmaf(w, h1.w, a1.w * cold);
        m = mn;
    }

    if (lane == 0) { m_lds[wave] = m; z_lds[wave] = z; }
    __syncthreads();

    // ---- combine the 8 wave-partials ----
    float M = -INFINITY;
    #pragma unroll
    for (int w = 0; w < WAVES; ++w) M = fmaxf(M, m_lds[w]);
    float Zg = 0.0f;
    #pragma unroll
    for (int w = 0; w < WAVES; ++w) Zg += z_lds[w] * __expf(m_lds[w] - M);
    const float invZ = 1.0f / Zg;

    // fold this wave's accumulator into shared hbar (ds_add_f32 atomics)
    const float f = __expf(m - M) * invZ;
    atomicAdd(&hbar[lane * 8 + 0], a0.x * f);
    atomicAdd(&hbar[lane * 8 + 1], a0.y * f);
    atomicAdd(&hbar[lane * 8 + 2], a0.z * f);
    atomicAdd(&hbar[lane * 8 + 3], a0.w * f);
    atomicAdd(&hbar[lane * 8 + 4], a1.x * f);
    atomicAdd(&hbar[lane * 8 + 5], a1.y * f);
    atomicAdd(&hbar[lane * 8 + 6], a1.z * f);
    atomicAdd(&hbar[lane * 8 + 7], a1.w * f);

    // alpha output: two rows per thread
    #pragma unroll
    for (int rep = 0; rep < NN / 256; ++rep) {
        const int n = t + rep * 256;
        alpha_out[(size_t)b * NN + n] = __expf(s_lds[n] - M) * invZ; // exp(-inf)=0
    }
    __syncthreads();

    hbar_out[(size_t)b * DG + t] = hbar[t];
}

// ---------------------------------------------------------------------------
// Kernel 2: ctx[b,e] = sum_d hbar[b,d] * Wv[e,d]
// 256x256x256 f32 GEMM via V_WMMA_F32_16X16X4_F32; one wave per 16x16 tile.
// ---------------------------------------------------------------------------
__global__ __launch_bounds__(256) void ctx_gemm_wmma(
    const float* __restrict__ Hbar,  // [B, DG]
    const float* __restrict__ Wv,    // [DD, DG]
    float* __restrict__ ctx)         // [B, DD]
{
    const int t    = threadIdx.x;
    const int lane = t & 31;
    const int wave = t >> 5;
    const int tile = blockIdx.x * WAVES + wave;   // 0..255
    const int m0   = (tile >> 4) << 4;            // batch-row block
    const int e0   = (tile & 15) << 4;            // desc-col block

    const int half = lane >> 4;                   // 0: K pair {0,1}, 1: {2,3}
    const int lr   = lane & 15;

    v8f c = {0.f, 0.f, 0.f, 0.f, 0.f, 0.f, 0.f, 0.f};

    #pragma unroll 4
    for (int k0 = 0; k0 < DG; k0 += 4) {
        // A tile 16x4 f32: lane<16 -> M=lr, K=k0+{0,1}; lane>=16 -> M=lr, K=k0+2+{0,1}
        const float* ap = Hbar + (size_t)(m0 + lr) * DG + k0 + half * 2;
        v2f a; a.x = ap[0]; a.y = ap[1];
        // B tile 4x16 f32 (= Wv^T slice): B[k][n] = Wv[e0+n][k0+k]
        const float* bp = Wv + (size_t)(e0 + lr) * DG + k0 + half * 2;
        v2f bm; bm.x = bp[0]; bm.y = bp[1];
        c = __builtin_amdgcn_wmma_f32_16x16x4_f32(
                /*neg_a=*/false, a, /*neg_b=*/false, bm,
                /*c_mod=*/(short)0, c, /*reuse_a=*/false, /*reuse_b=*/false);
    }

    // D layout: lane<16 -> M=r, N=lane ; lane>=16 -> M=r+8, N=lane-16
    #pragma unroll
    for (int r = 0; r < 8; ++r) {
        const int Mi = r + 8 * half;
        ctx[(size_t)(m0 + Mi) * DD + (e0 + lr)] = c[r];
    }
}

// ---------------------------------------------------------------------------
extern "C" void kernel_launch(void* const* d_in, const int* in_sizes, int n_in,
                              void* d_out, int out_size, void* d_ws, size_t ws_size,
                              hipStream_t stream) {
    const float*         tokns = (const float*)d_in[0];
    const float*         H     = (const float*)d_in[1];
    const unsigned char* mask  = (const unsigned char*)d_in[2];
    const float*         Wq    = (const float*)d_in[3];
    const float*         Wk    = (const float*)d_in[4];
    const float*         Wv    = (const float*)d_in[5];
    const float*         ls    = (const float*)d_in[6];

    float* out   = (float*)d_out;
    float* alpha = out;                       // [B, N]
    float* ctx   = out + (size_t)BB * NN;     // [B, DD]
    float* hbar  = (float*)d_ws;              // [B, DG] scratch (256 KB)

    attn_stream_kernel<<<dim3(BB), dim3(256), 0, stream>>>(
        tokns, H, mask, Wq, Wk, ls, alpha, hbar);
    ctx_gemm_wmma<<<dim3(BB / WAVES), dim3(256), 0, stream>>>(hbar, Wv, ctx);

    (void)in_sizes; (void)n_in; (void)out_size; (void)ws_size;
}